// RotMat_27290222199269
// MI455X (gfx1250) — compile-verified
//
#include <hip/hip_runtime.h>

typedef __attribute__((ext_vector_type(2))) float v2f;
typedef __attribute__((ext_vector_type(8))) float v8f;

#define NROWS    512
#define BCOLS    1024
#define SLABCOLS 8
#define GROUPS   8
#define RPG      64          // rounds per group (last group gets 63)
#define TOTAL_R  (NROWS - 1)

#define KC       32          // K-chunk staged in LDS per block
#define APSTR    17          // A LDS row stride in float2 (conflict-free: 17r mod 32 distinct)
#define BPSTR    80          // B LDS pair-row stride in float2 (>=64, ==16 mod 32)

// Circle-method round-robin schedule, closed form (player 0 fixed, others rotate).
__device__ __forceinline__ int rr_partner(int r, int n) {
  const int Nm1 = NROWS - 1;
  if (n == 0) {
    return ((Nm1 - 1 - r) % Nm1) + 1;
  }
  int pos = ((n - 1 + r) % Nm1) + 1;
  int q = (NROWS - 1) - pos;
  if (q == 0) return 0;
  int t = (q - 1 - r) % Nm1;
  if (t < 0) t += Nm1;
  return t + 1;
}

// ---------------------------------------------------------------------------
// Build partial product P_g = G_{r1-1} ... G_{r0} applied to identity.
// grid = (64 column-blocks of 8 cols, 8 groups), block = 256 (8 waves).
// ---------------------------------------------------------------------------
__global__ void __launch_bounds__(256) RotBuildP(const float* __restrict__ thetas,
                                                 float* __restrict__ Pout) {
  __shared__ float slab[NROWS * SLABCOLS];   // 16 KB column slab
  __shared__ float cS[NROWS];
  __shared__ float sS[NROWS];
  __shared__ short pS[NROWS];

  const int g    = blockIdx.y;
  const int c0   = blockIdx.x * SLABCOLS;
  const int tid  = threadIdx.x;
  const int lane = tid & 31;
  const int wv   = tid >> 5;
  const int col  = lane & 7;
  const int rsub = lane >> 3;
  const int rbase = wv * 4 + rsub;            // rows = rbase + 32*k (conflict-free)

#pragma unroll
  for (int k = 0; k < 16; ++k) {
    int row = rbase + 32 * k;
    slab[row * SLABCOLS + col] = (row == c0 + col) ? 1.0f : 0.0f;
  }
  __syncthreads();

  const int r0 = g * RPG;
  const int r1 = (r0 + RPG < TOTAL_R) ? (r0 + RPG) : TOTAL_R;

  for (int r = r0; r < r1; ++r) {
#pragma unroll
    for (int rr = 0; rr < 2; ++rr) {
      int row = tid * 2 + rr;
      int p = rr_partner(r, row);
      int i = row < p ? row : p;
      int j = row < p ? p : row;
      int t = i * (NROWS - 1) - (i * (i - 1)) / 2 + (j - i - 1);
      float th = thetas[t];
      float sn = __sinf(th);
      cS[row] = __cosf(th);
      sS[row] = (row == i) ? -sn : sn;
      pS[row] = (short)p;
    }
    __syncthreads();
    float nv[16];
#pragma unroll
    for (int k = 0; k < 16; ++k) {
      int row = rbase + 32 * k;
      int p = pS[row];
      nv[k] = cS[row] * slab[row * SLABCOLS + col] +
              sS[row] * slab[p   * SLABCOLS + col];
    }
    __syncthreads();
#pragma unroll
    for (int k = 0; k < 16; ++k) {
      int row = rbase + 32 * k;
      slab[row * SLABCOLS + col] = nv[k];
    }
    __syncthreads();
  }

  float* P = Pout + (size_t)g * NROWS * NROWS;
#pragma unroll
  for (int k = 0; k < 16; ++k) {
    int row = rbase + 32 * k;
    P[(size_t)row * NROWS + c0 + col] = slab[row * SLABCOLS + col];
  }
}

// ---------------------------------------------------------------------------
// Async stagers (ASYNCcnt-tracked). LDS byte offset = low 32 bits of the
// generic LDS pointer (flat aperture rule), used as the async VDST.
// A chunk: 128 rows x KC floats, K-pair interleaved: f2[row*APSTR + k/2].
// B chunk:  KC rows x 64 floats, K-pair interleaved: f2[(row/2)*BPSTR + col],
//           component row&1 -> fragment {k,k+1} is one aligned b64.
// ---------------------------------------------------------------------------
__device__ __forceinline__ void async_load_A(const float* gsrc, unsigned lds_base, int tid) {
#pragma unroll
  for (int p = 0; p < 8; ++p) {
    int e   = tid + p * 256;                 // 2048 K-pairs
    int kp  = e & 15;                        // pair within chunk
    int row = e >> 4;                        // 0..127
    const float* g = gsrc + (size_t)row * NROWS + 2 * kp;
    unsigned l = lds_base + (unsigned)(row * APSTR + kp) * 8u;
    asm volatile("global_load_async_to_lds_b64 %0, %1, off"
                 :: "v"(l), "v"(g) : "memory");
  }
}

__device__ __forceinline__ void async_load_B(const float* gsrc, unsigned lds_base,
                                             int ncols, int tid) {
#pragma unroll
  for (int p = 0; p < 8; ++p) {
    int e   = tid + p * 256;                 // 2048 floats
    int col = e & 63;
    int row = e >> 6;                        // 0..31
    const float* g = gsrc + (size_t)row * ncols + col;
    unsigned l = lds_base +
                 (unsigned)((((row >> 1) * BPSTR + col) << 3) + ((row & 1) << 2));
    asm volatile("global_load_async_to_lds_b32 %0, %1, off"
                 :: "v"(l), "v"(g) : "memory");
  }
}

// ---------------------------------------------------------------------------
// C = A @ B, M=K=512, N=ncols. Block = 8 waves -> 128x64 C tile; each wave a
// 16x64 strip (4 subtiles share one A fragment). Both A and B double-buffered
// in LDS via async-to-LDS; all hot-loop operand loads are single ds_load_b64.
// combine=1: batched pairwise product C[z] = src[2z+1] @ src[2z].
// ---------------------------------------------------------------------------
__global__ void __launch_bounds__(256) RotGemmWMMA(const float* __restrict__ Abase,
                                                   const float* __restrict__ Bbase,
                                                   float* __restrict__ Cbase,
                                                   int ncols, int combine) {
  constexpr int K = NROWS;
  constexpr size_t MS = (size_t)NROWS * NROWS;

  __shared__ v2f aTile[2][128 * APSTR];      // 2 x 17408 B
  __shared__ v2f bTile[2][(KC / 2) * BPSTR]; // 2 x 10240 B   (total 55296 B)

  const float* A = Abase;
  const float* B = Bbase;
  float*       C = Cbase;
  if (combine) {
    int mat = blockIdx.z;
    A = Abase + (size_t)(2 * mat + 1) * MS;
    B = Bbase + (size_t)(2 * mat) * MS;
    C = Cbase + (size_t)mat * MS;
  }

  const int tid  = threadIdx.x;
  const int lane = tid & 31;
  const int wv   = tid >> 5;
  const int half = lane >> 4;
  const int l16  = lane & 15;

  const int m0blk = blockIdx.x * 128;
  const int m0    = m0blk + wv * 16;
  const int n0blk = blockIdx.y * 64;

  const float* ablk = A + (size_t)m0blk * K;   // A[m0blk][0]
  const float* bblk = B + n0blk;               // B[0][n0blk]

  const unsigned ldsA0 = (unsigned)(uintptr_t)(&aTile[0][0]);
  const unsigned ldsA1 = (unsigned)(uintptr_t)(&aTile[1][0]);
  const unsigned ldsB0 = (unsigned)(uintptr_t)(&bTile[0][0]);
  const unsigned ldsB1 = (unsigned)(uintptr_t)(&bTile[1][0]);

  v8f acc0 = {}, acc1 = {}, acc2 = {}, acc3 = {};

  constexpr int kChunks = K / KC;              // 16

  // prologue: chunk 0 -> buffer 0
  async_load_A(ablk, ldsA0, tid);
  async_load_B(bblk, ldsB0, ncols, tid);
  asm volatile("s_wait_asynccnt 0x0" ::: "memory");
  __syncthreads();

  const int aRowBase = (wv * 16 + l16) * APSTR;  // + pair index

  for (int c = 0; c < kChunks; ++c) {
    const int cur = c & 1;
    const int kc0 = c * KC;

    if (c + 1 < kChunks) {                     // stream next chunk into other buffer
      async_load_A(ablk + (kc0 + KC), cur ? ldsA0 : ldsA1, tid);
      async_load_B(bblk + (size_t)(kc0 + KC) * ncols, cur ? ldsB0 : ldsB1, ncols, tid);
    }

    const v2f* aT = &aTile[cur][0];
    const v2f* bT = &bTile[cur][0];
#pragma unroll
    for (int kp = 0; kp < KC / 2; kp += 2) {   // kp = k/2; fragment covers k..k+3
      v2f a  = aT[aRowBase + kp + half];       // {A[m][k+2h], A[m][k+2h+1]}
      const int bb = (kp + half) * BPSTR + l16;
      v2f b0 = bT[bb];
      v2f b1 = bT[bb + 16];
      v2f b2 = bT[bb + 32];
      v2f b3 = bT[bb + 48];
      acc0 = __builtin_amdgcn_wmma_f32_16x16x4_f32(false, a, false, b0, (short)0, acc0, false, false);
      acc1 = __builtin_amdgcn_wmma_f32_16x16x4_f32(false, a, false, b1, (short)0, acc1, false, false);
      acc2 = __builtin_amdgcn_wmma_f32_16x16x4_f32(false, a, false, b2, (short)0, acc2, false, false);
      acc3 = __builtin_amdgcn_wmma_f32_16x16x4_f32(false, a, false, b3, (short)0, acc3, false, false);
    }

    if (c + 1 < kChunks) {
      asm volatile("s_wait_asynccnt 0x0" ::: "memory");
    }
    __syncthreads();                           // all waves done with cur; next chunk visible
  }

  // C/D layout: VGPR v -> row v + 8*half, col = lane%16
#pragma unroll
  for (int v = 0; v < 8; ++v) {
    float* crow = C + (size_t)(m0 + v + 8 * half) * ncols + n0blk + l16;
    crow[0]  = acc0[v];
    crow[16] = acc1[v];
    crow[32] = acc2[v];
    crow[48] = acc3[v];
  }
}

extern "C" void kernel_launch(void* const* d_in, const int* in_sizes, int n_in,
                              void* d_out, int out_size, void* d_ws, size_t ws_size,
                              hipStream_t stream) {
  const float* x      = (const float*)d_in[0];   // [512, 1024]
  const float* thetas = (const float*)d_in[1];   // [130816]
  float* out = (float*)d_out;                    // [512, 1024]

  constexpr size_t MS = (size_t)NROWS * NROWS;
  float* arena0 = (float*)d_ws;                  // 8 matrices (8 MB)
  float* arena1 = arena0 + (size_t)GROUPS * MS;  // 4 matrices (4 MB) -> 12 MB ws

  // 1) 8 partial products in parallel (serial depth 64 rounds instead of 511)
  RotBuildP<<<dim3(NROWS / SLABCOLS, GROUPS), 256, 0, stream>>>(thetas, arena0);
  // 2) binary-tree combine with f32 WMMA: 8 -> 4 -> 2 -> 1
  RotGemmWMMA<<<dim3(4, 8, 4), 256, 0, stream>>>(arena0, arena0, arena1, NROWS, 1);
  RotGemmWMMA<<<dim3(4, 8, 2), 256, 0, stream>>>(arena1, arena1, arena0, NROWS, 1);
  RotGemmWMMA<<<dim3(4, 8, 1), 256, 0, stream>>>(arena0, arena0, arena1, NROWS, 1);
  // 3) out = Q @ x  (Q = arena1 slot 0)
  RotGemmWMMA<<<dim3(4, 16, 1), 256, 0, stream>>>(arena1, x, out, BCOLS, 0);
}